// FluxPosRegressor_85323820302814
// MI455X (gfx1250) — compile-verified
//
#include <hip/hip_runtime.h>
#include <stdint.h>

// Problem geometry from the reference: N=256 images, H=W=512, all f32.
#define H_DIM 512
#define W_DIM 512

typedef float v4f __attribute__((ext_vector_type(4)));  // native vector for NT store

__device__ __forceinline__ int mirror_idx(int idx, int n) {
    // period = 2*(n-1); python-style mod (non-negative), then reflect
    const int p = 2 * (n - 1);
    int m = idx % p;
    if (m < 0) m += p;
    if (m >= n) m = p - m;
    return m;
}

__global__ __launch_bounds__(128)
void FluxPosRegressor_shift_kernel(const float* __restrict__ inputs,
                                   const float* __restrict__ flux,
                                   const float* __restrict__ noise,
                                   const float* __restrict__ dxdy,
                                   float* __restrict__ out) {
    // Two staged source rows: 2 * 512 floats = 4 KB of the 320 KB/WGP LDS.
    __shared__ float lds_rows[2 * W_DIM];

    const int y   = blockIdx.x;   // output row
    const int n   = blockIdx.y;   // image index
    const int tid = threadIdx.x;  // 0..127, each handles 4 columns

    // Per-image scalars (block-uniform -> scalar loads).
    const float dx = dxdy[2 * (size_t)n + 0];
    const float dy = dxdy[2 * (size_t)n + 1];
    const float fx = flux[n];
    const float nz = noise[n];

    // Row interpolation parameters, computed exactly like the reference
    // (f32 add then floor, so rounding matches bit-for-bit).
    const float r   = (float)y + dy;
    const float r0f = floorf(r);
    const float tr  = r - r0f;
    const int   r0  = (int)r0f;
    const int   mr0 = mirror_idx(r0,     H_DIM);
    const int   mr1 = mirror_idx(r0 + 1, H_DIM);

    // Async global -> LDS staging of the two source rows.
    // Each of 128 lanes moves 16 bytes per row (b128): 128*16 = 2048 B = one row.
    const float* src0 = inputs + ((size_t)n * H_DIM + mr0) * W_DIM + tid * 4;
    const float* src1 = inputs + ((size_t)n * H_DIM + mr1) * W_DIM + tid * 4;

    // Low 32 bits of a generic pointer to __shared__ are the LDS byte address.
    unsigned lds0 = (unsigned)(uintptr_t)(&lds_rows[tid * 4]);
    unsigned lds1 = (unsigned)(uintptr_t)(&lds_rows[W_DIM + tid * 4]);
    unsigned long long ga0 = (unsigned long long)(uintptr_t)src0;
    unsigned long long ga1 = (unsigned long long)(uintptr_t)src1;

    asm volatile("global_load_async_to_lds_b128 %0, %1, off"
                 :: "v"(lds0), "v"(ga0) : "memory");
    asm volatile("global_load_async_to_lds_b128 %0, %1, off"
                 :: "v"(lds1), "v"(ga1) : "memory");
    asm volatile("s_wait_asynccnt 0" ::: "memory");
    __syncthreads();  // make all 4 waves' staged data visible

    // Compute 4 output columns per thread from LDS, one b128 NT store.
    v4f res;
    const int xbase = tid * 4;
#pragma unroll
    for (int j = 0; j < 4; ++j) {
        const int   x   = xbase + j;
        const float c   = (float)x + dx;   // reference f32 rounding
        const float c0f = floorf(c);
        const float tc  = c - c0f;
        const int   c0  = (int)c0f;
        const int   mc0 = mirror_idx(c0,     W_DIM);
        const int   mc1 = mirror_idx(c0 + 1, W_DIM);

        const float g00 = lds_rows[mc0];
        const float g01 = lds_rows[mc1];
        const float g10 = lds_rows[W_DIM + mc0];
        const float g11 = lds_rows[W_DIM + mc1];

        const float top = (1.0f - tc) * g00 + tc * g01;
        const float bot = (1.0f - tc) * g10 + tc * g11;
        const float v   = (1.0f - tr) * top + tr * bot;
        res[j] = v * fx + nz;
    }

    // Output is streamed (written once, never re-read): non-temporal store
    // keeps the 192 MB L2 for the input rows, whose adjacent-row reuse is
    // what holds HBM read traffic at 1x.
    v4f* dst = (v4f*)(out + ((size_t)n * H_DIM + y) * W_DIM + xbase);
    __builtin_nontemporal_store(res, dst);
}

extern "C" void kernel_launch(void* const* d_in, const int* in_sizes, int n_in,
                              void* d_out, int out_size, void* d_ws, size_t ws_size,
                              hipStream_t stream) {
    (void)n_in; (void)out_size; (void)d_ws; (void)ws_size;
    const float* inputs = (const float*)d_in[0];  // (1, N, H, W) f32
    const float* flux   = (const float*)d_in[1];  // (N, 1, 1)   f32
    const float* noise  = (const float*)d_in[2];  // (N, 1, 1)   f32
    const float* dxdy   = (const float*)d_in[3];  // (N, 2)      f32
    float*       out    = (float*)d_out;          // (N, H, W)   f32

    const int N = in_sizes[1];  // number of images (flux element count)

    dim3 grid(H_DIM, N);        // one block per (row, image)
    FluxPosRegressor_shift_kernel<<<grid, 128, 0, stream>>>(inputs, flux, noise, dxdy, out);
}